// BPR_19731079758339
// MI455X (gfx1250) — compile-verified
//
#include <hip/hip_runtime.h>
#include <hip/hip_bf16.h>
#include <cstddef>

// ---------------------------------------------------------------------------
// Problem constants (from reference module scope)
// ---------------------------------------------------------------------------
#define D_MODEL 512
#define DA      256
#define FACTOR  128
#define BS      4096
#define N_UNI   24576
#define P_U     98304
#define P_J     16384
// _seg_ids gives contiguous, uniform segments:
//   piece_seg: P_U/N_UNI = 4 pieces / unique text
//   user_seg : N_UNI/BS  = 6 texts  / user
//   itemj_seg: P_J/BS    = 4 pieces / item-j
#define R_PIECE 4
#define R_USER  6
#define R_ITEMJ 4

typedef __attribute__((ext_vector_type(2))) float v2f;
typedef __attribute__((ext_vector_type(8))) float v8f;

// ---------------------------------------------------------------------------
// Kernel 0: fold attention projections into 512-vectors:
//   v01[d] = sum_a ws02[a] * Ws01[a][d];  v1[d] = sum_a ws2[a] * Ws1[a][d]
// ---------------------------------------------------------------------------
__global__ __launch_bounds__(256) void attn_vec_kernel(
    const float* __restrict__ Ws01, const float* __restrict__ ws02,
    const float* __restrict__ Ws1,  const float* __restrict__ ws2,
    float* __restrict__ vout /* [0..511]=v01, [512..1023]=v1 */) {
  int d = blockIdx.x * blockDim.x + threadIdx.x;
  if (d < D_MODEL) {
    float a0 = 0.f, a1 = 0.f;
    for (int a = 0; a < DA; ++a) {
      a0 = fmaf(ws02[a], Ws01[a * D_MODEL + d], a0);
      a1 = fmaf(ws2[a],  Ws1[a * D_MODEL + d],  a1);
    }
    vout[d] = a0;
    vout[D_MODEL + d] = a1;
  }
}

// ---------------------------------------------------------------------------
// Kernels 1-3: fused segment attention pooling, one wave32 per segment.
// Lane l owns dims {4l..4l+3} + 128q, q=0..3 (float4, fully coalesced).
// Rows kept in registers: logit GEMV, wave softmax, weighted sum — one pass.
// ---------------------------------------------------------------------------
template <int R>
__global__ __launch_bounds__(256) void pool_seg_kernel(
    const float* __restrict__ X, const float* __restrict__ v,
    float* __restrict__ out, int nseg) {
  const int lane = threadIdx.x & 31;
  const int wave = threadIdx.x >> 5;
  const int seg  = blockIdx.x * 8 + wave;
  if (seg >= nseg) return;

  const float* base = X + (size_t)seg * R * D_MODEL;
  // speculative prefetch of the next block-stride segment (global_prefetch_b8)
  __builtin_prefetch(base + (size_t)8 * R * D_MODEL + lane * 4, 0, 0);

  float4 vv[4];
#pragma unroll
  for (int q = 0; q < 4; ++q)
    vv[q] = *(const float4*)(v + lane * 4 + q * 128);

  float4 xr[R][4];
  float  s[R];
#pragma unroll
  for (int r = 0; r < R; ++r) {
    float p = 0.f;
#pragma unroll
    for (int q = 0; q < 4; ++q) {
      float4 x = *(const float4*)(base + r * D_MODEL + lane * 4 + q * 128);
      xr[r][q] = x;
      p = fmaf(x.x, vv[q].x, p);
      p = fmaf(x.y, vv[q].y, p);
      p = fmaf(x.z, vv[q].z, p);
      p = fmaf(x.w, vv[q].w, p);
    }
#pragma unroll
    for (int m = 16; m >= 1; m >>= 1) p += __shfl_xor(p, m, 32);
    s[r] = p;  // full logit, replicated in all 32 lanes
  }

  float mx = s[0];
#pragma unroll
  for (int r = 1; r < R; ++r) mx = fmaxf(mx, s[r]);
  float a[R], z = 0.f;
#pragma unroll
  for (int r = 0; r < R; ++r) { a[r] = __expf(s[r] - mx); z += a[r]; }
  const float invz = 1.0f / z;

  float* o = out + (size_t)seg * D_MODEL;
#pragma unroll
  for (int q = 0; q < 4; ++q) {
    float4 acc = {0.f, 0.f, 0.f, 0.f};
#pragma unroll
    for (int r = 0; r < R; ++r) {
      const float w = a[r] * invz;
      acc.x = fmaf(xr[r][q].x, w, acc.x);
      acc.y = fmaf(xr[r][q].y, w, acc.y);
      acc.z = fmaf(xr[r][q].z, w, acc.z);
      acc.w = fmaf(xr[r][q].w, w, acc.w);
    }
    *(float4*)(o + lane * 4 + q * 128) = acc;
  }
}

// ---------------------------------------------------------------------------
// Kernel 4: itemi_rep = uni_user_vec[itemi_pos : itemi_pos+BS]  (pos on device)
// ---------------------------------------------------------------------------
__global__ __launch_bounds__(256) void slice_copy_kernel(
    const float* __restrict__ uni, const int* __restrict__ pos,
    float* __restrict__ out) {
  const size_t i = (size_t)blockIdx.x * blockDim.x + threadIdx.x;
  const float4* s4 = (const float4*)(uni + (size_t)(*pos) * D_MODEL);
  float4* o4 = (float4*)out;
  if (i < (size_t)BS * D_MODEL / 4) o4[i] = s4[i];
}

// ---------------------------------------------------------------------------
// Kernel 5: BPR scoring with V_WMMA_F32_16X16X4_F32.
// Per wave: 16 rows of M. For each of 8 N-tiles, run three K=512 WMMA chains
// (user_lat / i_lat / j_lat), bias folded into the C-init, then accumulate
// pred partials in-register. D-layout: lane%16 = N col, VGPR r = M row
// r + 8*(lane/16) -> row-sum reduction = shfl_xor over masks {8,4,2,1}.
// ---------------------------------------------------------------------------
__global__ __launch_bounds__(256) void bpr_score_wmma_kernel(
    const float* __restrict__ userRep,  const float* __restrict__ itemiRep,
    const float* __restrict__ itemjRep, const float* __restrict__ userW,
    const float* __restrict__ userB,    const float* __restrict__ itemW,
    const float* __restrict__ itemB,    float* __restrict__ predI,
    float* __restrict__ predJ) {
  const int lane = threadIdx.x & 31;
  const int wave = threadIdx.x >> 5;
  const int lo   = lane & 15;   // N within tile / row within A
  const int hi   = lane >> 4;   // K-pair select within WMMA operand
  const int m0   = (blockIdx.x * 8 + wave) * 16;  // wave's 16 output rows

  const float* ru = userRep  + (size_t)(m0 + lo) * D_MODEL;
  const float* ri = itemiRep + (size_t)(m0 + lo) * D_MODEL;
  const float* rj = itemjRep + (size_t)(m0 + lo) * D_MODEL;

  float pi[8] = {0.f, 0.f, 0.f, 0.f, 0.f, 0.f, 0.f, 0.f};
  float pj[8] = {0.f, 0.f, 0.f, 0.f, 0.f, 0.f, 0.f, 0.f};

  for (int nt = 0; nt < FACTOR / 16; ++nt) {
    const int n = nt * 16 + lo;
    const float ub = userB[n], ib = itemB[n];
    v8f au = {ub, ub, ub, ub, ub, ub, ub, ub};  // bias as C-init
    v8f ai = {ib, ib, ib, ib, ib, ib, ib, ib};
    v8f aj = ai;
    const float* wu = userW + (size_t)n * D_MODEL;
    const float* wi = itemW + (size_t)n * D_MODEL;

    for (int k = 0; k < D_MODEL; k += 4) {
      const int ko = k + 2 * hi;                 // A/B operand K-slot per lane
      v2f a_u = *(const v2f*)(ru + ko);
      v2f b_u = *(const v2f*)(wu + ko);          // B[k][n] = W[n][k]
      au = __builtin_amdgcn_wmma_f32_16x16x4_f32(
          false, a_u, false, b_u, (short)0, au, false, false);
      v2f a_i = *(const v2f*)(ri + ko);
      v2f b_i = *(const v2f*)(wi + ko);
      ai = __builtin_amdgcn_wmma_f32_16x16x4_f32(
          false, a_i, false, b_i, (short)0, ai, false, false);
      v2f a_j = *(const v2f*)(rj + ko);
      aj = __builtin_amdgcn_wmma_f32_16x16x4_f32(
          false, a_j, false, b_i, (short)0, aj, false, false);
    }
#pragma unroll
    for (int r = 0; r < 8; ++r) {
      pi[r] = fmaf(au[r], ai[r], pi[r]);
      pj[r] = fmaf(au[r], aj[r], pj[r]);
    }
  }

  // reduce over the 16 N-lanes of each half-wave (stays within hi group)
#pragma unroll
  for (int r = 0; r < 8; ++r) {
#pragma unroll
    for (int m = 8; m >= 1; m >>= 1) {
      pi[r] += __shfl_xor(pi[r], m, 32);
      pj[r] += __shfl_xor(pj[r], m, 32);
    }
  }
  if (lo == 0) {
#pragma unroll
    for (int r = 0; r < 8; ++r) {
      predI[m0 + hi * 8 + r] = pi[r];
      predJ[m0 + hi * 8 + r] = pj[r];
    }
  }
}

// ---------------------------------------------------------------------------
// Launch
// ---------------------------------------------------------------------------
extern "C" void kernel_launch(void* const* d_in, const int* in_sizes, int n_in,
                              void* d_out, int out_size, void* d_ws, size_t ws_size,
                              hipStream_t stream) {
  const float* user_pooled = (const float*)d_in[0];   // [P_U, 512]
  const float* itemj_piece = (const float*)d_in[1];   // [P_J, 512]
  const float* Ws1         = (const float*)d_in[2];   // [256, 512]
  const float* ws2         = (const float*)d_in[3];   // [1, 256]
  const float* Ws01        = (const float*)d_in[4];   // [256, 512]
  const float* ws02        = (const float*)d_in[5];   // [1, 256]
  const float* user_W      = (const float*)d_in[6];   // [128, 512]
  const float* user_b      = (const float*)d_in[7];   // [128]
  const float* item_W      = (const float*)d_in[8];   // [128, 512]
  const float* item_b      = (const float*)d_in[9];   // [128]
  const int*   itemi_pos   = (const int*)d_in[13];    // scalar

  // workspace layout
  float* v01      = (float*)d_ws;                       // 512
  float* v1       = v01 + D_MODEL;                      // 512
  float* uni      = v1 + D_MODEL;                       // [N_UNI, 512]
  float* user_rep = uni + (size_t)N_UNI * D_MODEL;      // [BS, 512]

  // output layout (tuple flattened in return order)
  float* itemi_rep = (float*)d_out;                     // [BS, 512]
  float* itemj_rep = itemi_rep + (size_t)BS * D_MODEL;  // [BS, 512]
  float* pred_i    = itemj_rep + (size_t)BS * D_MODEL;  // [BS]
  float* pred_j    = pred_i + BS;                       // [BS]

  // 0) fold attention weight vectors
  attn_vec_kernel<<<2, 256, 0, stream>>>(Ws01, ws02, Ws1, ws2, v01);
  // 1) pieces -> unique user texts (4 pieces/segment)
  pool_seg_kernel<R_PIECE><<<N_UNI / 8, 256, 0, stream>>>(user_pooled, v01, uni, N_UNI);
  // 2) unique texts -> users (6 texts/segment)
  pool_seg_kernel<R_USER><<<BS / 8, 256, 0, stream>>>(uni, v1, user_rep, BS);
  // 3) item-j pieces -> item-j reps (4 pieces/segment), straight into output
  pool_seg_kernel<R_ITEMJ><<<BS / 8, 256, 0, stream>>>(itemj_piece, v01, itemj_rep, BS);
  // 4) itemi_rep = uni[itemi_pos : itemi_pos+BS]
  slice_copy_kernel<<<(BS * D_MODEL / 4 + 255) / 256, 256, 0, stream>>>(
      uni, itemi_pos, itemi_rep);
  // 5) fused WMMA latent projections + BPR scores
  bpr_score_wmma_kernel<<<BS / (8 * 16), 256, 0, stream>>>(
      user_rep, itemi_rep, itemj_rep, user_W, user_b, item_W, item_b,
      pred_i, pred_j);
}